// LatticeGraphSAGE_28527172780199
// MI455X (gfx1250) — compile-verified
//
#include <hip/hip_runtime.h>

#define NEG_SLOPE 0.01f

typedef float v2f __attribute__((ext_vector_type(2)));
typedef float v8f __attribute__((ext_vector_type(8)));

// ---------------------------------------------------------------- utilities
__global__ void zero_kernel(float* __restrict__ p, long long n) {
  long long i = (long long)blockIdx.x * blockDim.x + threadIdx.x;
  long long stride = (long long)gridDim.x * blockDim.x;
  for (; i < n; i += stride) p[i] = 0.0f;
}

__global__ void degree_kernel(const long long* __restrict__ dst,
                              float* __restrict__ deg, int E) {
  int e = blockIdx.x * blockDim.x + threadIdx.x;
  if (e < E) atomicAdd(&deg[(int)dst[e]], 1.0f);
}

__global__ void rdeg_kernel(const float* __restrict__ deg,
                            float* __restrict__ rdeg, int n) {
  int i = blockIdx.x * blockDim.x + threadIdx.x;
  if (i < n) rdeg[i] = 1.0f / fmaxf(deg[i], 1.0f);
}

// ------------------------------------------------------- edge scatter (agg)
// Half-wave (16 lanes) per edge: lanes cooperatively stream one full 64-float
// row (coalesced 256B) and scatter-add it with f32 global atomics (L2 atomic
// units; table is L2-resident at 25.6MB << 192MB).
__global__ void scatter_kernel(const float* __restrict__ h,
                               const long long* __restrict__ src,
                               const long long* __restrict__ dst,
                               float* __restrict__ agg, int E) {
  long long gid = (long long)blockIdx.x * blockDim.x + threadIdx.x;
  int e = (int)(gid >> 4);
  int c = (int)(gid & 15);
  if (e >= E) return;
  int s = (int)src[e];
  int d = (int)dst[e];
  const float4 v = *(const float4*)(h + (long long)s * 64 + c * 4);
  float* p = agg + (long long)d * 64 + c * 4;
  atomicAdd(p + 0, v.x);
  atomicAdd(p + 1, v.y);
  atomicAdd(p + 2, v.z);
  atomicAdd(p + 3, v.w);
}

// ---------------------------------------------- fused SAGE linear (fp32 WMMA)
// h_out = leakyrelu( (agg*rdeg) @ Wl^T + bl + h_in @ Wr^T )
// One wave32 per 16-row tile; 4 waves per block. Weights staged in LDS with
// stride-66 padding (even stride -> 8B-aligned pairs, conflict-free banks).
// A frag (16x4 f32): lane holds row m = base + (lane&15), K = 2*(lane>>4)+{0,1}
// B frag (4x16 f32): lane holds col n, same K pair; B[k][n] == W[n][k].
// C/D (16x16 f32): VGPR r <-> row r + 8*(lane>>4), col = lane&15.
__global__ __launch_bounds__(128)
void sage_layer_kernel(const float* __restrict__ agg,
                       const float* __restrict__ rdeg,
                       const float* __restrict__ hin,
                       const float* __restrict__ Wl,
                       const float* __restrict__ bl,
                       const float* __restrict__ Wr,
                       float* __restrict__ hout, int ntiles) {
  __shared__ float sWl[64 * 66];
  __shared__ float sWr[64 * 66];
  const int tid = threadIdx.x;
  for (int i = tid; i < 64 * 64; i += 128) {
    int r = i >> 6, c = i & 63;
    sWl[r * 66 + c] = Wl[i];
    sWr[r * 66 + c] = Wr[i];
  }
  __syncthreads();

  const int wave = tid >> 5;
  const int lane = tid & 31;
  const int tile = blockIdx.x * 4 + wave;
  if (tile >= ntiles) return;

  const int half = lane >> 4;   // which K pair / which row half of C
  const int lid  = lane & 15;   // row (A) or col (B,C) within 16
  const int kb   = half * 2;
  const int m    = tile * 16 + lid;

  const float rd = rdeg[m];
  const float* aggRow = agg + (long long)m * 64 + kb;
  const float* hinRow = hin + (long long)m * 64 + kb;

  v8f acc0 = {}, acc1 = {}, acc2 = {}, acc3 = {};

#pragma unroll
  for (int kk = 0; kk < 16; ++kk) {
    const int k0 = kk * 4;
    v2f am, as;
    {
      const float2 t = *(const float2*)(aggRow + k0);
      am.x = t.x * rd; am.y = t.y * rd;          // mean-aggregated neighbors
    }
    {
      const float2 t = *(const float2*)(hinRow + k0);
      as.x = t.x; as.y = t.y;                    // self features
    }
#pragma unroll
    for (int nt = 0; nt < 4; ++nt) {
      const int n = nt * 16 + lid;
      const float* wl = &sWl[n * 66 + k0 + kb];
      const float* wr = &sWr[n * 66 + k0 + kb];
      v2f bL; bL.x = wl[0]; bL.y = wl[1];
      v2f bR; bR.x = wr[0]; bR.y = wr[1];
      v8f* accp = (nt == 0) ? &acc0 : (nt == 1) ? &acc1 : (nt == 2) ? &acc2 : &acc3;
      *accp = __builtin_amdgcn_wmma_f32_16x16x4_f32(false, am, false, bL,
                                                    (short)0, *accp, false, false);
      *accp = __builtin_amdgcn_wmma_f32_16x16x4_f32(false, as, false, bR,
                                                    (short)0, *accp, false, false);
    }
  }

  // bias + leaky-relu + store
  v8f accs[4] = {acc0, acc1, acc2, acc3};
#pragma unroll
  for (int nt = 0; nt < 4; ++nt) {
    const int col = nt * 16 + lid;
    const float bias = bl[col];
#pragma unroll
    for (int r = 0; r < 8; ++r) {
      const int row = tile * 16 + r + half * 8;
      float v = accs[nt][r] + bias;
      v = (v >= 0.0f) ? v : NEG_SLOPE * v;
      hout[(long long)row * 64 + col] = v;
    }
  }
}

// ------------------------------------------------------------ final 64 -> 1
__global__ void out_kernel(const float* __restrict__ h,
                           const float* __restrict__ Wout,
                           const float* __restrict__ bout,
                           float* __restrict__ out, int n) {
  int i = blockIdx.x * blockDim.x + threadIdx.x;
  if (i >= n) return;
  const float* row = h + (long long)i * 64;
  float s = bout[0];
#pragma unroll
  for (int k = 0; k < 64; ++k) s += row[k] * Wout[k];
  out[i] = s;
}

// ----------------------------------------------------------------- launcher
extern "C" void kernel_launch(void* const* d_in, const int* in_sizes, int n_in,
                              void* d_out, int out_size, void* d_ws, size_t ws_size,
                              hipStream_t stream) {
  const float*     x    = (const float*)d_in[0];
  const long long* ei   = (const long long*)d_in[1];
  const float* Wl[3] = {(const float*)d_in[2], (const float*)d_in[5], (const float*)d_in[8]};
  const float* bl[3] = {(const float*)d_in[3], (const float*)d_in[6], (const float*)d_in[9]};
  const float* Wr[3] = {(const float*)d_in[4], (const float*)d_in[7], (const float*)d_in[10]};
  const float* Wout = (const float*)d_in[11];
  const float* bout = (const float*)d_in[12];

  const int N = in_sizes[0] / 64;
  const int E = in_sizes[1] / 2;
  const long long* src = ei;
  const long long* dstv = ei + E;

  float* agg  = (float*)d_ws;
  float* hA   = agg + (size_t)N * 64;
  float* hB   = hA  + (size_t)N * 64;
  float* deg  = hB  + (size_t)N * 64;
  float* rdeg = deg + N;

  const int ntiles = (N + 15) / 16;

  // degrees (layer-invariant)
  zero_kernel<<<256, 256, 0, stream>>>(deg, (long long)N);
  degree_kernel<<<(E + 255) / 256, 256, 0, stream>>>(dstv, deg, E);
  rdeg_kernel<<<(N + 255) / 256, 256, 0, stream>>>(deg, rdeg, N);

  const float* hin = x;
  float* houts[3] = {hA, hB, hA};
  for (int l = 0; l < 3; ++l) {
    zero_kernel<<<2048, 256, 0, stream>>>(agg, (long long)N * 64);
    long long t = (long long)E * 16;
    scatter_kernel<<<(int)((t + 255) / 256), 256, 0, stream>>>(hin, src, dstv, agg, E);
    sage_layer_kernel<<<(ntiles + 3) / 4, 128, 0, stream>>>(
        agg, rdeg, hin, Wl[l], bl[l], Wr[l], houts[l], ntiles);
    hin = houts[l];
  }
  out_kernel<<<(N + 255) / 256, 256, 0, stream>>>(hin, Wout, bout, (float*)d_out, N);
}